// GCNLayer_1657857376311
// MI455X (gfx1250) — compile-verified
//
#include <hip/hip_runtime.h>
#include <hip/hip_bf16.h>

#define N_FEATS 64
#define WTP_LD  80   // float2 stride; 80*8B = 640B ≡ 128B (mod 256B) -> halves hit disjoint bank sets

typedef float v2f __attribute__((ext_vector_type(2)));
typedef float v8f __attribute__((ext_vector_type(8)));

// ---------------------------------------------------------------- zero init
__global__ void zero_f32(float* __restrict__ p, int n) {
    int i = blockIdx.x * blockDim.x + threadIdx.x;
    if (i < n) p[i] = 0.0f;
}

// ------------------------------------------------------------- edge scatter
// 16 lanes per edge, float4 per lane: one full 256B feature row per edge,
// coalesced gather from x[src], fp32 atomic scatter into agg[dst] (L2-resident).
__global__ void edge_scatter(const float* __restrict__ x,
                             const int*   __restrict__ src,
                             const int*   __restrict__ dst,
                             float*       __restrict__ agg,
                             float*       __restrict__ deg,
                             int n_edges) {
    int t = blockIdx.x * blockDim.x + threadIdx.x;
    int e = t >> 4;
    if (e >= n_edges) return;
    int l = t & 15;
    int s = src[e];
    int d = dst[e];
    const float4 v = *reinterpret_cast<const float4*>(x + (size_t)s * N_FEATS + l * 4);
    float* o = agg + (size_t)d * N_FEATS + l * 4;
    atomicAdd(o + 0, v.x);
    atomicAdd(o + 1, v.y);
    atomicAdd(o + 2, v.z);
    atomicAdd(o + 3, v.w);
    if (l == 0) atomicAdd(deg + d, 1.0f);
}

// ------------------------------------------- fused select + WMMA f32 GEMM
// out[m][n] = (deg[m]>0 ? agg[m] : x[m]) @ W^T + b
// One wave -> one 16-row M tile x all 64 output features (4 accumulators),
// K=64 via 16 chained V_WMMA_F32_16X16X4_F32 per accumulator.
// W^T is staged in LDS packed as K-pairs so each B fragment is one ds_load_b64
// landing in an even-aligned VGPR pair (no v_mov marshalling before v_wmma).
__global__ void __launch_bounds__(256)
gcn_linear_wmma(const float* __restrict__ x,
                const float* __restrict__ agg,
                const float* __restrict__ deg,
                const float* __restrict__ W,     // [64 out][64 in], row-major
                const float* __restrict__ bias,  // [64]
                float*       __restrict__ out,
                int n_tiles) {
    // WtP[kp][n] = { W[n][2kp], W[n][2kp+1] }  (i.e. a K-pair of W^T column n)
    __shared__ float WtPf[32 * WTP_LD * 2];
    __shared__ float bsh[N_FEATS];

    // Stage packed W^T into LDS (coalesced global reads of W row-major).
    for (int i = threadIdx.x; i < N_FEATS * N_FEATS; i += blockDim.x) {
        int n = i / N_FEATS;       // output feature (column of W^T)
        int k = i % N_FEATS;       // input feature  (row of W^T)
        WtPf[((k >> 1) * WTP_LD + n) * 2 + (k & 1)] = W[i];
    }
    if (threadIdx.x < N_FEATS) bsh[threadIdx.x] = bias[threadIdx.x];
    __syncthreads();

    const int wid  = threadIdx.x >> 5;
    const int lane = threadIdx.x & 31;
    const int tile = blockIdx.x * 8 + wid;          // wave-uniform
    if (tile >= n_tiles) return;                    // whole wave exits; EXEC stays all-1s

    const int half = lane >> 4;                     // 0: K pair {4s,4s+1}, 1: {4s+2,4s+3}
    const int l16  = lane & 15;
    const int row  = tile * 16 + l16;               // A-matrix row for this lane

    const bool use_agg = deg[row] > 0.0f;
    const float* hrow = use_agg ? (agg + (size_t)row * N_FEATS)
                                : (x   + (size_t)row * N_FEATS);

    v8f acc0 = {}, acc1 = {}, acc2 = {}, acc3 = {};

#pragma unroll
    for (int s = 0; s < 16; ++s) {
        const int kp = 2 * s + half;    // this lane's K-pair index (A and B use same split)

        // A fragment: lane holds h[row][2kp], h[row][2kp+1]
        v2f a = *reinterpret_cast<const v2f*>(hrow + 2 * kp);

        // B fragments: one contiguous float2 per N-tile from packed LDS
        const v2f* wrow = reinterpret_cast<const v2f*>(&WtPf[kp * WTP_LD * 2]);
        v2f b0 = wrow[l16 +  0];
        v2f b1 = wrow[l16 + 16];
        v2f b2 = wrow[l16 + 32];
        v2f b3 = wrow[l16 + 48];

        acc0 = __builtin_amdgcn_wmma_f32_16x16x4_f32(false, a, false, b0, (short)0, acc0, false, false);
        acc1 = __builtin_amdgcn_wmma_f32_16x16x4_f32(false, a, false, b1, (short)0, acc1, false, false);
        acc2 = __builtin_amdgcn_wmma_f32_16x16x4_f32(false, a, false, b2, (short)0, acc2, false, false);
        acc3 = __builtin_amdgcn_wmma_f32_16x16x4_f32(false, a, false, b3, (short)0, acc3, false, false);
    }

    // D layout: VGPR g -> (M = g + 8*half, N = l16 + 16*ntile)
    float* obase = out + (size_t)(tile * 16 + 8 * half) * N_FEATS + l16;
#pragma unroll
    for (int g = 0; g < 8; ++g) {
        float* orow = obase + (size_t)g * N_FEATS;
        orow[ 0] = acc0[g] + bsh[l16 +  0];
        orow[16] = acc1[g] + bsh[l16 + 16];
        orow[32] = acc2[g] + bsh[l16 + 32];
        orow[48] = acc3[g] + bsh[l16 + 48];
    }
}

// ------------------------------------------------------------------ launch
extern "C" void kernel_launch(void* const* d_in, const int* in_sizes, int n_in,
                              void* d_out, int out_size, void* d_ws, size_t ws_size,
                              hipStream_t stream) {
    const float* x    = (const float*)d_in[0];  // [N, 64]
    const int*   src  = (const int*)  d_in[1];  // [E]
    const int*   dst  = (const int*)  d_in[2];  // [E]
    const float* W    = (const float*)d_in[3];  // [64, 64]
    const float* bias = (const float*)d_in[4];  // [64]
    float*       out  = (float*)d_out;          // [N, 64]

    const int n_nodes = in_sizes[0] / N_FEATS;  // 100000 (divisible by 16)
    const int n_edges = in_sizes[1];            // 1600000

    // Workspace: agg [N,64] f32, then deg [N] f32.
    float* agg = (float*)d_ws;
    float* deg = agg + (size_t)n_nodes * N_FEATS;

    // 1) zero agg + deg
    const int n_zero = n_nodes * N_FEATS + n_nodes;
    zero_f32<<<(n_zero + 255) / 256, 256, 0, stream>>>(agg, n_zero);

    // 2) edge scatter (16 threads per edge)
    const int scat_threads = n_edges * 16;
    edge_scatter<<<(scat_threads + 255) / 256, 256, 0, stream>>>(
        x, src, dst, agg, deg, n_edges);

    // 3) fused select + WMMA linear
    const int n_tiles = (n_nodes + 15) / 16;    // 6250
    const int blocks  = (n_tiles + 7) / 8;      // 8 waves (M tiles) per block
    gcn_linear_wmma<<<blocks, 256, 0, stream>>>(x, agg, deg, W, bias, out, n_tiles);
}